// SWTDecomposition_3599182594827
// MI455X (gfx1250) — compile-verified
//
#include <hip/hip_runtime.h>
#include <cstdint>

#define TLEN     4096
#define NTHREADS 256
#define KPT      (TLEN / NTHREADS)   // 16 samples per thread

typedef __attribute__((ext_vector_type(4))) float        v4f;
typedef __attribute__((ext_vector_type(4))) unsigned int u32x4;
typedef __attribute__((ext_vector_type(8))) int          i32x8;
typedef __attribute__((ext_vector_type(4))) int          i32x4;

// db4 dec_lo / dec_hi, pywt swt(norm=True) => scaled by 1/sqrt(2).
// hi[k] = (-1)^(7-k) * lo[7-k]
#define S2_ 0.70710678118654752440084436210485
__device__ constexpr float LO[8] = {
    (float)(-0.010597401784997278 * S2_),
    (float)( 0.032883011666982945 * S2_),
    (float)( 0.030841381835986965 * S2_),
    (float)(-0.18703481171888114  * S2_),
    (float)(-0.02798376941698385  * S2_),
    (float)( 0.6308807679295904   * S2_),
    (float)( 0.7148465705525415   * S2_),
    (float)( 0.23037781330885523  * S2_),
};
__device__ constexpr float HI[8] = {
    (float)(-0.23037781330885523  * S2_),
    (float)( 0.7148465705525415   * S2_),
    (float)(-0.6308807679295904   * S2_),
    (float)(-0.02798376941698385  * S2_),
    (float)( 0.18703481171888114  * S2_),
    (float)( 0.030841381835986965 * S2_),
    (float)(-0.032883011666982945 * S2_),
    (float)(-0.010597401784997278 * S2_),
};

// Periodized a-trous 8-tap conv: out[t] = sum_j f[j] * buf[(t + D*(4-j)) mod T]
template <int D>
__device__ __forceinline__ void conv8(const float* __restrict__ buf, int t,
                                      float& outA, float& outD) {
    float a = 0.f, d = 0.f;
#pragma unroll
    for (int j = 0; j < 8; ++j) {
        float v = buf[(t + D * (4 - j)) & (TLEN - 1)];
        a = fmaf(LO[j], v, a);
        d = fmaf(HI[j], v, d);
    }
    outA = a;
    outD = d;
}

__global__ __launch_bounds__(NTHREADS)
void swt_db4_tdm_kernel(const float* __restrict__ x, float* __restrict__ out) {
    __shared__ __align__(16) float sx[TLEN];  // input row
    __shared__ __align__(16) float a1[TLEN];  // cA level 1
    __shared__ __align__(16) float a2[TLEN];  // cA level 2

    const int row = blockIdx.x;
    const int tid = threadIdx.x;
    const float* __restrict__ xr = x + (size_t)row * TLEN;

#if defined(__gfx1250__) && __has_builtin(__builtin_amdgcn_tensor_load_to_lds)
    // --- Tensor Data Mover: DMA one 16KB row (1x4096 fp32 tile) into LDS ---
    if (tid < 32) {  // single wave issues the TDM op (EXEC ignored by TDM)
        const unsigned long long ga  = (unsigned long long)(uintptr_t)xr;
        const unsigned int       lds = (unsigned int)(uintptr_t)&sx[0];

        u32x4 g0;
        g0.x = 1u;                                     // count=1 (valid D#), user mode
        g0.y = lds;                                    // lds_addr (bytes)
        g0.z = (unsigned int)ga;                       // global_addr[31:0]
        g0.w = (unsigned int)((ga >> 32) & 0x01FFFFFFull) | (2u << 30); // addr[56:32] | type=2

        i32x8 g1;
        g1[0] = (int)(2u << 16);                       // data_size = 4 bytes
        g1[1] = (int)(((unsigned)TLEN & 0xFFFFu) << 16); // tensor_dim0[15:0] @ bits63:48
        g1[2] = (int)((((unsigned)TLEN >> 16) & 0xFFFFu) | (1u << 16)); // dim0 hi | tensor_dim1=1
        g1[3] = (int)(((unsigned)TLEN & 0xFFFFu) << 16); // tile_dim0 = 4096
        g1[4] = 1;                                     // tile_dim1 = 1, tile_dim2 = 0
        g1[5] = TLEN;                                  // tensor_dim0_stride[31:0]
        g1[6] = 0;                                     // stride0 hi | stride1 lo
        g1[7] = 0;

        i32x4 gz4 = {0, 0, 0, 0};                      // groups 2/3 unused (2D tensor)
        i32x8 gz8 = {0, 0, 0, 0, 0, 0, 0, 0};          // extra group (clang-23 6-arg form)
        __builtin_amdgcn_tensor_load_to_lds(g0, g1, gz4, gz4, gz8, 0);
        __builtin_amdgcn_s_wait_tensorcnt(0);
    }
    __syncthreads();
#else
    // Fallback (also keeps the host-compile pass clean): vectorized coop load.
    {
        v4f* dst = (v4f*)sx;
        const v4f* src = (const v4f*)xr;
#pragma unroll
        for (int k = 0; k < TLEN / 4 / NTHREADS; ++k)
            dst[tid + k * NTHREADS] = src[tid + k * NTHREADS];
    }
    __syncthreads();
#endif

    float d1[KPT], d2[KPT], d3[KPT], a3[KPT];

    // Level 1: d=1, sx -> a1 (LDS), cD1 stays in registers
#pragma unroll
    for (int k = 0; k < KPT; ++k) {
        const int t = tid + k * NTHREADS;
        float cA;
        conv8<1>(sx, t, cA, d1[k]);
        a1[t] = cA;
    }
    __syncthreads();

    // Level 2: d=2, a1 -> a2 (LDS), cD2 in registers
#pragma unroll
    for (int k = 0; k < KPT; ++k) {
        const int t = tid + k * NTHREADS;
        float cA;
        conv8<2>(a1, t, cA, d2[k]);
        a2[t] = cA;
    }
    __syncthreads();

    // Level 3: d=4, a2 -> registers only
#pragma unroll
    for (int k = 0; k < KPT; ++k) {
        const int t = tid + k * NTHREADS;
        conv8<4>(a2, t, a3[k], d3[k]);
    }

    // Streamed output, never re-read: non-temporal 128-bit coalesced stores.
    // Channel order per reference: [cA3, cD3, cD2, cD1]
    v4f* __restrict__ o4 = (v4f*)(out + (size_t)row * TLEN * 4);
#pragma unroll
    for (int k = 0; k < KPT; ++k) {
        const int t = tid + k * NTHREADS;
        v4f v;
        v.x = a3[k];
        v.y = d3[k];
        v.z = d2[k];
        v.w = d1[k];
        __builtin_nontemporal_store(v, &o4[t]);
    }
}

extern "C" void kernel_launch(void* const* d_in, const int* in_sizes, int n_in,
                              void* d_out, int out_size, void* d_ws, size_t ws_size,
                              hipStream_t stream) {
    const float* x = (const float*)d_in[0];
    float* out = (float*)d_out;
    const int rows = in_sizes[0] / TLEN;  // 64*64 = 4096 independent rows
    swt_db4_tdm_kernel<<<rows, NTHREADS, 0, stream>>>(x, out);
}